// nBRCLayer_55654186221616
// MI455X (gfx1250) — compile-verified
//
#include <hip/hip_runtime.h>
#include <hip/hip_bf16.h>

// nBRC layer for MI455X (gfx1250, wave32, WMMA).
//
// The 512-step scan is the critical path: per-step work is tiny (~0.5 GFLOP)
// and the whole working set (weights 3.5MB + x_seq 128MB) is L2-resident
// (192MB L2), so we use EXACT fp32 matrix math (V_WMMA_F32_16X16X4_F32) and
// optimize for *latency*: with only 512 waves/step (~1 wave per SIMD32) there
// is no occupancy to hide L2 latency, so the K-loops are software-pipelined
// with a depth-8 rotating register buffer (32 b64 loads in flight) and fully
// unrolled so every load uses an immediate IOFFSET instead of VALU address
// arithmetic. VGPR pressure is irrelevant at this occupancy.

#define T_LEN 512
#define BATCH 256
#define ISZ   256
#define HSZ   512

typedef float v2f __attribute__((ext_vector_type(2)));
typedef float v8f __attribute__((ext_vector_type(8)));

static __device__ __forceinline__ v2f ld2(const float* p) {
    return *(const v2f*)p;
}

// D = A(16x4) * B(4x16) + C, all fp32, wave32 (ISA 7.12.2 layouts):
//  A: lane L holds row M=L%16; v0/v1 = K=(L/16)*2 + {0,1}
//  B: lane L holds col N=L%16; v0/v1 = K=(L/16)*2 + {0,1}
//  C/D: lane L holds col N=L%16; VGPR v holds row M = v + (L/16)*8
static __device__ __forceinline__ v8f wmma_f32(v2f a, v2f b, v8f c) {
    return __builtin_amdgcn_wmma_f32_16x16x4_f32(
        /*neg_a=*/false, a, /*neg_b=*/false, b,
        /*c_mod=*/(short)0, c, /*reuse_a=*/false, /*reuse_b=*/false);
}

// One scan step: h_out = c*h + (1-c)*tanh(xh + a*h),
//   c = sigmoid(x U_c^T + b_c + h W_c^T), a = 1 + tanh(x U_a^T + b_a + h W_a^T),
//   xh = x U_h^T + b_h.
// Grid: 128 blocks x 128 threads (4 waves); one wave owns one 16x16 tile of
// the [BATCH x HSZ] output (512 tiles).
__global__ void __launch_bounds__(128)
nbrc_step_kernel(const float* __restrict__ x_t,      // [BATCH, ISZ]
                 const float* __restrict__ h_prev,   // [BATCH, HSZ]
                 const float* __restrict__ U_c,      // [HSZ, ISZ]
                 const float* __restrict__ W_c,      // [HSZ, HSZ]
                 const float* __restrict__ b_c,      // [HSZ]
                 const float* __restrict__ U_a,
                 const float* __restrict__ W_a,
                 const float* __restrict__ b_a,
                 const float* __restrict__ U_h,
                 const float* __restrict__ b_h,
                 float* __restrict__ h_out)          // [BATCH, HSZ]
{
    const int wid  = threadIdx.x >> 5;            // wave in block (0..3)
    const int lane = threadIdx.x & 31;
    const int tile = blockIdx.x * 4 + wid;        // 0..511
    const int m0   = (tile & 15) << 4;            // batch-tile origin
    const int n0   = (tile >> 4) << 4;            // hidden-tile origin

    const int n  = lane & 15;                     // row (A) / col (B,C,D) in tile
    const int kp = (lane >> 4) << 1;              // K sub-offset {0,2}

    // Per-lane strip bases (A rows use lane%16 just like B cols).
    const float* xr  = x_t    + (size_t)(m0 + n) * ISZ + kp;
    const float* hr  = h_prev + (size_t)(m0 + n) * HSZ + kp;
    const float* ucr = U_c    + (size_t)(n0 + n) * ISZ + kp;
    const float* uar = U_a    + (size_t)(n0 + n) * ISZ + kp;
    const float* uhr = U_h    + (size_t)(n0 + n) * ISZ + kp;
    const float* wcr = W_c    + (size_t)(n0 + n) * HSZ + kp;
    const float* war = W_a    + (size_t)(n0 + n) * HSZ + kp;

    v8f accC = {}, accA = {}, accH = {};   // input projections (K = ISZ)
    v8f accHC = {}, accHA = {};            // recurrent terms   (K = HSZ)

    constexpr int PF = 8;                  // prefetch depth (K-groups in flight)

    // --- Phase 1: xc/xa/xh = x_t @ U^T, K = 256 -----------------------------
    {
        constexpr int NG = ISZ / 4;        // 64 K-groups
        v2f a[PF], p1[PF], p2[PF], p3[PF];
#pragma unroll
        for (int i = 0; i < PF; ++i) {
            a[i]  = ld2(xr  + 4 * i);
            p1[i] = ld2(ucr + 4 * i);
            p2[i] = ld2(uar + 4 * i);
            p3[i] = ld2(uhr + 4 * i);
        }
#pragma unroll
        for (int g = 0; g < NG; ++g) {
            const int s = g & (PF - 1);
            const v2f av = a[s], v1 = p1[s], v2 = p2[s], v3 = p3[s];
            if (g + PF < NG) {             // refill slot PF groups ahead
                const int kn = (g + PF) * 4;
                a[s]  = ld2(xr  + kn);
                p1[s] = ld2(ucr + kn);
                p2[s] = ld2(uar + kn);
                p3[s] = ld2(uhr + kn);
            }
            accC = wmma_f32(av, v1, accC);
            accA = wmma_f32(av, v2, accA);
            accH = wmma_f32(av, v3, accH);
        }
    }

    // --- Phase 2: hc/ha = h_prev @ W^T, K = 512 -----------------------------
    {
        constexpr int NG = HSZ / 4;        // 128 K-groups
        v2f a[PF], p1[PF], p2[PF];
#pragma unroll
        for (int i = 0; i < PF; ++i) {
            a[i]  = ld2(hr  + 4 * i);
            p1[i] = ld2(wcr + 4 * i);
            p2[i] = ld2(war + 4 * i);
        }
#pragma unroll
        for (int g = 0; g < NG; ++g) {
            const int s = g & (PF - 1);
            const v2f av = a[s], v1 = p1[s], v2 = p2[s];
            if (g + PF < NG) {
                const int kn = (g + PF) * 4;
                a[s]  = ld2(hr  + kn);
                p1[s] = ld2(wcr + kn);
                p2[s] = ld2(war + kn);
            }
            accHC = wmma_f32(av, v1, accHC);
            accHA = wmma_f32(av, v2, accHA);
        }
    }

    // --- Gate nonlinearity + write h_t --------------------------------------
    const float bcv = b_c[n0 + n];
    const float bav = b_a[n0 + n];
    const float bhv = b_h[n0 + n];
    const int mofs = (lane >> 4) << 3;            // C/D row offset {0,8}

#pragma unroll
    for (int v = 0; v < 8; ++v) {
        const int m = m0 + v + mofs;
        const size_t idx = (size_t)m * HSZ + (n0 + n);
        const float hp = h_prev[idx];
        const float cg = 1.0f / (1.0f + expf(-(accC[v] + bcv + accHC[v])));
        const float ag = 1.0f + tanhf(accA[v] + bav + accHA[v]);
        const float hn = cg * hp + (1.0f - cg) * tanhf(accH[v] + bhv + ag * hp);
        h_out[idx] = hn;
    }
}

extern "C" void kernel_launch(void* const* d_in, const int* in_sizes, int n_in,
                              void* d_out, int out_size, void* d_ws, size_t ws_size,
                              hipStream_t stream) {
    const float* x_seq = (const float*)d_in[0];   // [T, B, I]
    const float* h0    = (const float*)d_in[1];   // [B, H]
    const float* U_c   = (const float*)d_in[2];
    const float* W_c   = (const float*)d_in[3];
    const float* b_c   = (const float*)d_in[4];
    const float* U_a   = (const float*)d_in[5];
    const float* W_a   = (const float*)d_in[6];
    const float* b_a   = (const float*)d_in[7];
    const float* U_h   = (const float*)d_in[8];
    const float* b_h   = (const float*)d_in[9];

    float* y = (float*)d_out;                     // y_seq [T,B,H] then h_final [B,H]
    const size_t stepElems = (size_t)BATCH * HSZ;

    for (int t = 0; t < T_LEN; ++t) {
        const float* h_prev = (t == 0) ? h0 : (y + (size_t)(t - 1) * stepElems);
        nbrc_step_kernel<<<dim3(128), dim3(128), 0, stream>>>(
            x_seq + (size_t)t * BATCH * ISZ, h_prev,
            U_c, W_c, b_c, U_a, W_a, b_a, U_h, b_h,
            y + (size_t)t * stepElems);
    }

    // h_final = y_seq[T-1]  (graph-capture-safe async D2D copy)
    hipMemcpyAsync(y + (size_t)T_LEN * stepElems,
                   y + (size_t)(T_LEN - 1) * stepElems,
                   stepElems * sizeof(float), hipMemcpyDeviceToDevice, stream);
}